// GaussianKernels_66219805770169
// MI455X (gfx1250) — compile-verified
//
#include <hip/hip_runtime.h>
#include <hip/hip_bf16.h>
#include <stdint.h>

#define NUM_CLASSES 100
#define NUM_NEIGH   200
#define NUM_CENTRES 100000
#define DIM         128
#define BATCH       1024
#define GC          0.005f            // 1/(2*sigma^2), sigma=10
#define HIST_BINS   8192              // key>>18 for finite non-negative f32 => < 8160
#define CAND_CAP    4096
#define N_TILES     (NUM_CENTRES/16)  // 6250
#define M_SUPER     (BATCH/64)        // 16 supertiles of 64 rows (4 waves x 16)
#define GEMM_GRID_X 512

typedef float v2f __attribute__((ext_vector_type(2)));
typedef float v8f __attribute__((ext_vector_type(8)));

// ---------------- K1: centre squared norms ----------------
__global__ __launch_bounds__(256) void centre_norms_kernel(
    const float* __restrict__ centres, float* __restrict__ c2) {
  int n = blockIdx.x * 256 + threadIdx.x;
  if (n >= NUM_CENTRES) return;
  const float4* row = (const float4*)(centres + (size_t)n * DIM);
  float acc = 0.f;
#pragma unroll
  for (int i = 0; i < DIM / 4; ++i) {
    float4 v = row[i];
    acc += v.x * v.x + v.y * v.y + v.z * v.z + v.w * v.w;
  }
  c2[n] = acc;
}

// ---------------- K2/K4: WMMA distance GEMM ----------------
// 4 waves per block, each owns a 16-row M-tile; the 16x128 centre tile is
// shared through LDS, double-buffered via async global->LDS copies.
// MODE 0: per-row histogram of fp32 keys (key>>18).
// MODE 1: collect (key,index) candidates with bin <= selbin[row].
template <int MODE>
__global__ __launch_bounds__(128) void gemm_pass_kernel(
    const float* __restrict__ features, const float* __restrict__ centres,
    const float* __restrict__ c2,
    uint32_t* __restrict__ hist,
    const uint32_t* __restrict__ selbin,
    unsigned long long* __restrict__ cand, uint32_t* __restrict__ cand_cnt) {
  __shared__ float sB[2][16][132];   // padded rows: conflict-free ds_load_b64
  const int tid  = threadIdx.x;
  const int lane = tid & 31;
  const int wave = tid >> 5;
  const int hi    = lane >> 4;   // 0/1
  const int lo    = lane & 15;
  const int kLane = 2 * hi;      // K offset within group of 4
  const int mbase = blockIdx.y * 64 + wave * 16;

  // ---- A fragments (16x4 f32 layout) live in registers for the whole kernel
  v2f aF[32];
  {
    const float* frow = features + (size_t)(mbase + lo) * DIM + kLane;
    float f2p = 0.f;
#pragma unroll
    for (int k = 0; k < 32; ++k) {
      float2 t = *(const float2*)(frow + 4 * k);
      aF[k].x = t.x; aF[k].y = t.y;
      f2p += t.x * t.x + t.y * t.y;
    }
    // fold the two half-K partials: lane lo now holds ||features[mbase+lo]||^2
    aF[0].x += 0.f;  // keep clang from reordering the shfl above the loads
    float f2v = f2p + __shfl_xor(f2p, 16, 32);
    // stash in a register-resident variable
    aF[31].x += 0.f;
    // (f2v used below via shfl per output row)
    // ---- async staging setup: thread copies 64B of the tile (4 x b128)
    const int ldr  = tid >> 3;        // row 0..15
    const int lseg = (tid & 7) * 16;  // starting float within the row
    const uint32_t ldsOff0 = (uint32_t)(uintptr_t)&sB[0][ldr][lseg];
    const uint32_t ldsOff1 = (uint32_t)(uintptr_t)&sB[1][ldr][lseg];

    auto stage = [&](int buf, int nt) {
      const float* g = centres + (size_t)(nt * 16 + ldr) * DIM + lseg;
      uint32_t l = buf ? ldsOff1 : ldsOff0;
#pragma unroll
      for (int q = 0; q < 4; ++q) {
        const float* gq = g + q * 4;
        uint32_t lq = l + q * 16;
        asm volatile("global_load_async_to_lds_b128 %0, %1, off"
                     :: "v"(lq), "v"(gq)
                     : "memory");
      }
    };

    const int nt0 = blockIdx.x;
    if (nt0 < N_TILES) stage(0, nt0);

    int p = 0;
    for (int nt = nt0; nt < N_TILES; nt += gridDim.x, ++p) {
      const int buf = p & 1;
      asm volatile("s_wait_asynccnt 0x0" ::: "memory");
      __syncthreads();  // tile `buf` fully in LDS; prev reads of buf^1 done
      const int ntn = nt + gridDim.x;
      if (ntn < N_TILES) stage(buf ^ 1, ntn);  // overlaps with compute below

      v8f acc = {};
#pragma unroll
      for (int k = 0; k < 32; ++k) {
        const float2 bv = *(const float2*)&sB[buf][lo][4 * k + kLane];
        v2f b; b.x = bv.x; b.y = bv.y;
        acc = __builtin_amdgcn_wmma_f32_16x16x4_f32(
            false, aF[k], false, b, (short)0, acc, false, false);
      }

      const int nbase = nt * 16;
      const float c2v = c2[nbase + lo];
#pragma unroll
      for (int v = 0; v < 8; ++v) {
        const int m = v + 8 * hi;               // D layout: M = v + 8*(lane>=16)
        const float fm = __shfl(f2v, v + 8 * hi, 32);
        float d2 = fm + c2v - 2.0f * acc[v];    // N = lane%16
        d2 = fmaxf(d2, 0.0f);
        uint32_t key = __float_as_uint(d2);     // order-preserving for d2 >= 0
        uint32_t bin = key >> 18;
        if (bin > HIST_BINS - 1) bin = HIST_BINS - 1;
        const int row = mbase + m;
        if (MODE == 0) {
          atomicAdd(&hist[(size_t)row * HIST_BINS + bin], 1u);
        } else {
          if (bin <= selbin[row]) {
            uint32_t pos = atomicAdd(&cand_cnt[row], 1u);
            if (pos < CAND_CAP) {
              cand[(size_t)row * CAND_CAP + pos] =
                  ((unsigned long long)key << 32) | (uint32_t)(nbase + lo);
            }
          }
        }
      }
      __syncthreads();  // all waves done reading sB[buf] before next overwrite
    }
  }
}

// ---------------- K3: per-row find bin containing the 200th smallest --------
__global__ __launch_bounds__(256) void select_bin_kernel(
    const uint32_t* __restrict__ hist, uint32_t* __restrict__ selbin) {
  __shared__ uint32_t part[256];
  const int row = blockIdx.x;
  const int tid = threadIdx.x;
  const uint32_t* h = hist + (size_t)row * HIST_BINS;
  const int per = HIST_BINS / 256;  // 32
  uint32_t s = 0;
  for (int i = 0; i < per; ++i) s += h[tid * per + i];
  part[tid] = s;
  __syncthreads();
  if (tid == 0) {
    uint32_t run = 0;
    int seg = 255;
    for (int t = 0; t < 256; ++t) {
      if (run + part[t] >= NUM_NEIGH) { seg = t; break; }
      run += part[t];
    }
    uint32_t b = (uint32_t)seg * per;
    for (; b < (uint32_t)(seg + 1) * per; ++b) {
      run += h[b];
      if (run >= NUM_NEIGH) break;
    }
    selbin[row] = b;
  }
}

// ---------------- K5: per-row bitonic sort of candidates -> top-200 ---------
__global__ __launch_bounds__(256) void topk_kernel(
    const unsigned long long* __restrict__ cand,
    const uint32_t* __restrict__ cand_cnt, uint32_t* __restrict__ nb_idx) {
  __shared__ unsigned long long s[CAND_CAP];  // 32 KB LDS
  const int row = blockIdx.x;
  const int tid = threadIdx.x;
  uint32_t cnt = cand_cnt[row];
  if (cnt > CAND_CAP) cnt = CAND_CAP;
  for (int i = tid; i < CAND_CAP; i += 256)
    s[i] = (i < (int)cnt) ? cand[(size_t)row * CAND_CAP + i] : ~0ULL;
  for (int k = 2; k <= CAND_CAP; k <<= 1)
    for (int j = k >> 1; j > 0; j >>= 1) {
      __syncthreads();
      for (int i = tid; i < CAND_CAP; i += 256) {
        int ixj = i ^ j;
        if (ixj > i) {
          unsigned long long a = s[i], b = s[ixj];
          bool up = ((i & k) == 0);
          if ((a > b) == up) { s[i] = b; s[ixj] = a; }
        }
      }
    }
  __syncthreads();
  if (tid < NUM_NEIGH)
    nb_idx[(size_t)row * NUM_NEIGH + tid] = (uint32_t)s[tid];
}

// ---------------- K6: rescore neighbours, class scatter, log-softmax --------
__global__ __launch_bounds__(256) void classify_kernel(
    const float* __restrict__ features, const float* __restrict__ centres,
    const float* __restrict__ weight, const long long* __restrict__ labels,
    const uint32_t* __restrict__ nb_idx, float* __restrict__ out) {
  __shared__ float p[NUM_CLASSES];
  __shared__ float sf[DIM];
  __shared__ float stot;
  const int row  = blockIdx.x;
  const int tid  = threadIdx.x;
  const int lane = tid & 31;
  const int wave = tid >> 5;
  if (tid < NUM_CLASSES) p[tid] = 0.f;
  if (tid < DIM) sf[tid] = features[(size_t)row * DIM + tid];
  __syncthreads();
  for (int k = wave; k < NUM_NEIGH; k += 8) {
    const uint32_t nb = nb_idx[(size_t)row * NUM_NEIGH + k];
    const float* __restrict__ c = centres + (size_t)nb * DIM;
    float acc = 0.f;
    for (int d = lane; d < DIM; d += 32) {
      float t = sf[d] - c[d];
      acc += t * t;
    }
#pragma unroll
    for (int off = 16; off > 0; off >>= 1) acc += __shfl_xor(acc, off, 32);
    if (lane == 0) {
      float w = expf(-acc * GC) * expf(weight[nb]);
      atomicAdd(&p[(int)labels[nb]], w);
    }
  }
  __syncthreads();
  if (tid == 0) {
    float tot = 0.f;
    for (int c = 0; c < NUM_CLASSES; ++c) {
      float v = p[c];
      if (v == 0.f) v = 1e-10f;
      p[c] = v;
      tot += v;
    }
    stot = tot;
  }
  __syncthreads();
  if (tid < NUM_CLASSES)
    out[(size_t)row * NUM_CLASSES + tid] = logf(p[tid] / stot);
}

// ---------------- host ----------------
extern "C" void kernel_launch(void* const* d_in, const int* in_sizes, int n_in,
                              void* d_out, int out_size, void* d_ws,
                              size_t ws_size, hipStream_t stream) {
  (void)in_sizes; (void)n_in; (void)out_size; (void)ws_size;
  const float* features     = (const float*)d_in[0];
  const float* centres      = (const float*)d_in[1];
  const float* weight       = (const float*)d_in[2];
  const long long* labels   = (const long long*)d_in[3];
  float* out = (float*)d_out;

  char* ws = (char*)d_ws;
  size_t off = 0;
  auto take = [&](size_t bytes) -> void* {
    void* p = ws + off;
    off = (off + bytes + 255) & ~(size_t)255;
    return p;
  };
  float*    c2       = (float*)take((size_t)NUM_CENTRES * sizeof(float));
  uint32_t* selbin   = (uint32_t*)take((size_t)BATCH * sizeof(uint32_t));
  uint32_t* cand_cnt = (uint32_t*)take((size_t)BATCH * sizeof(uint32_t));
  uint32_t* nb_idx   = (uint32_t*)take((size_t)BATCH * NUM_NEIGH * sizeof(uint32_t));
  uint32_t* hist     = (uint32_t*)take((size_t)BATCH * HIST_BINS * sizeof(uint32_t));
  unsigned long long* cand =
      (unsigned long long*)take((size_t)BATCH * CAND_CAP * sizeof(unsigned long long));

  hipMemsetAsync(hist, 0, (size_t)BATCH * HIST_BINS * sizeof(uint32_t), stream);
  hipMemsetAsync(cand_cnt, 0, (size_t)BATCH * sizeof(uint32_t), stream);

  centre_norms_kernel<<<(NUM_CENTRES + 255) / 256, 256, 0, stream>>>(centres, c2);

  dim3 ggrid(GEMM_GRID_X, M_SUPER);
  gemm_pass_kernel<0><<<ggrid, 128, 0, stream>>>(
      features, centres, c2, hist, nullptr, nullptr, nullptr);
  select_bin_kernel<<<BATCH, 256, 0, stream>>>(hist, selbin);
  gemm_pass_kernel<1><<<ggrid, 128, 0, stream>>>(
      features, centres, c2, nullptr, selbin, cand, cand_cnt);
  topk_kernel<<<BATCH, 256, 0, stream>>>(cand, cand_cnt, nb_idx);
  classify_kernel<<<BATCH, 256, 0, stream>>>(
      features, centres, weight, labels, nb_idx, out);
}